// CARAFENaive_72181220376847
// MI455X (gfx1250) — compile-verified
//
#include <hip/hip_runtime.h>
#include <hip/hip_bf16.h>

typedef __attribute__((ext_vector_type(2))) float v2f;
typedef __attribute__((ext_vector_type(8))) float v8f;

#define KSZ    5
#define PAD    2
#define C_TOT  256
#define H_IN   64
#define W_IN   64
#define H_OUT  128
#define W_OUT  128
#define KDIM   40        // 5 (di) * 8 (band cols)
#define NSTEP  10        // KDIM / 4 (K per wmma_f32_16x16x4)
#define LDSTR  44        // LDS row stride in floats (44c mod 64 distinct for c=0..15)

__global__ __launch_bounds__(256)
void carafe_wmma_f32(const float* __restrict__ feat,
                     const float* __restrict__ masks,
                     float* __restrict__ out) {
  __shared__ float ldsA[C_TOT * LDSTR];   // A panel: [channel][di*8+col]
  __shared__ float ldsB[16 * LDSTR];      // B band (transposed): [p][kk]

  const int g   = blockIdx.x;            // 8 * 64 * 16 groups
  const int n   = g >> 10;               // / (64*16)
  const int rem = g & 1023;
  const int h   = rem >> 4;              // low-res row
  const int w0  = (rem & 15) << 2;       // low-res col group base (4 wide)

  const int tid  = threadIdx.x;
  const int lane = tid & 31;
  const int wave = tid >> 5;

  // ---- Stage A panel: feat[n, :, h-2..h+2, w0-2..w0+5], zero-padded ----
  {
    const int c = tid;                   // 256 threads -> one channel each
    const float* fbase = feat + (size_t)(n * C_TOT + c) * (H_IN * W_IN);
    #pragma unroll
    for (int di = 0; di < KSZ; ++di) {
      const int row = h - PAD + di;
      float* dst = &ldsA[c * LDSTR + di * 8];
      if (row >= 0 && row < H_IN) {
        const float* src = fbase + row * W_IN + (w0 - PAD);
        #pragma unroll
        for (int j = 0; j < 4; ++j) {    // pairs are fully in/out (w0 % 4 == 0)
          const int col0 = w0 - PAD + 2 * j;
          v2f v = {0.f, 0.f};
          if (col0 >= 0 && col0 + 1 < W_IN)
            v = *(const v2f*)(src + 2 * j);   // 8B-aligned (col0 even)
          *(v2f*)(dst + 2 * j) = v;
        }
      } else {
        #pragma unroll
        for (int j = 0; j < 8; ++j) dst[j] = 0.f;
      }
    }
  }

  // ---- Build banded B: B[kk=(di,col)][p=(a,wloc)] = masks or 0 ----
  for (int idx = tid; idx < 16 * KDIM; idx += 256) {
    const int p    = idx / KDIM;
    const int kk   = idx - p * KDIM;
    const int di   = kk >> 3;
    const int col  = kk & 7;
    const int wloc = p & 7;
    const int a    = p >> 3;
    const int wl   = wloc >> 1;
    const int dj   = col - wl;           // band condition
    float v = 0.f;
    if (dj >= 0 && dj < KSZ) {
      const int i  = di * KSZ + dj;
      const int Ho = 2 * h + a;
      const int Wo = 2 * w0 + wloc;
      v = masks[(((size_t)n * (KSZ * KSZ) + i) * H_OUT + Ho) * W_OUT + Wo];
    }
    ldsB[p * LDSTR + kk] = v;
  }

  __syncthreads();

  // ---- WMMA compute: D[16c x 16px] = A[16c x 40] * B[40 x 16px] ----
  const int m16  = lane & 15;
  const int koff = (lane >> 4) << 1;     // K sub-offset: 0 (lanes 0-15) / 2 (16-31)

  v2f bfrag[NSTEP];
  #pragma unroll
  for (int s = 0; s < NSTEP; ++s)
    bfrag[s] = *(const v2f*)&ldsB[m16 * LDSTR + s * 4 + koff];

  const int wloc = m16 & 7;
  const int a    = m16 >> 3;
  const int Ho   = 2 * h + a;
  const int Wo   = 2 * w0 + wloc;

  #pragma unroll
  for (int t = 0; t < 2; ++t) {          // each wave: 2 channel tiles
    const int ct = wave + t * 8;
    v8f acc = {};
    #pragma unroll
    for (int s = 0; s < NSTEP; ++s) {
      v2f afrag = *(const v2f*)&ldsA[(ct * 16 + m16) * LDSTR + s * 4 + koff];
      acc = __builtin_amdgcn_wmma_f32_16x16x4_f32(
          /*neg_a=*/false, afrag, /*neg_b=*/false, bfrag[s],
          /*c_mod=*/(short)0, acc, /*reuse_a=*/false, /*reuse_b=*/false);
    }
    // D layout: VGPR r -> channel ct*16 + r (+8 for upper lane half), lane -> pixel
    const int cbase = ct * 16 + ((lane >> 4) << 3);
    float* obase = out + ((size_t)(n * C_TOT + cbase) * H_OUT + Ho) * W_OUT + Wo;
    #pragma unroll
    for (int r = 0; r < 8; ++r)
      __builtin_nontemporal_store(acc[r], obase + (size_t)r * (H_OUT * W_OUT));
  }
}

extern "C" void kernel_launch(void* const* d_in, const int* in_sizes, int n_in,
                              void* d_out, int out_size, void* d_ws, size_t ws_size,
                              hipStream_t stream) {
  (void)in_sizes; (void)n_in; (void)d_ws; (void)ws_size; (void)out_size;
  const float* feat  = (const float*)d_in[0];   // (8,256,64,64) fp32
  const float* masks = (const float*)d_in[1];   // (8,25,128,128) fp32
  float* out = (float*)d_out;                   // (8,256,128,128) fp32

  const int groups = 8 * 64 * (64 / 4);         // 8192 blocks
  carafe_wmma_f32<<<groups, 256, 0, stream>>>(feat, masks, out);
}